// LowRankBilinearAttention_20882130993372
// MI455X (gfx1250) — compile-verified
//
#include <hip/hip_runtime.h>
#include <math.h>

// ---------------------------------------------------------------------------
// Problem constants (from the reference)
// ---------------------------------------------------------------------------
#define BB   8
#define PP   196
#define LL   80
#define DIM1 2048
#define DIM2 300
#define ATT  1024

typedef __attribute__((ext_vector_type(2))) float v2f;
typedef __attribute__((ext_vector_type(8))) float v8f;

// Single-instruction tanh on gfx1250 (v_tanh_f32, TRANS op: co-executes with
// VALU). Fallback to ocml tanhf if the builtin is unavailable.
__device__ __forceinline__ float fast_tanh(float x) {
#if __has_builtin(__builtin_amdgcn_tanhf)
    return __builtin_amdgcn_tanhf(x);
#else
    return tanhf(x);
#endif
}

// ---------------------------------------------------------------------------
// v[a] = sum_c wt[c] * Wh[c, a]   (coalesced over a; Wh is ATT x ATT row-major)
// ---------------------------------------------------------------------------
__global__ void precompute_v_kernel(const float* __restrict__ Wh,
                                    const float* __restrict__ wt,
                                    float* __restrict__ vout) {
    int a = blockIdx.x * blockDim.x + threadIdx.x;   // 0..ATT-1
    float acc = 0.0f;
    #pragma unroll 4
    for (int c = 0; c < ATT; ++c)
        acc += wt[c] * Wh[(size_t)c * ATT + a];
    vout[a] = acc;
}

// ---------------------------------------------------------------------------
// c0 = dot(bh, wt) + bt
// ---------------------------------------------------------------------------
__global__ void precompute_c0_kernel(const float* __restrict__ bh,
                                     const float* __restrict__ wt,
                                     const float* __restrict__ bt,
                                     float* __restrict__ c0out) {
    __shared__ float s[256];
    float acc = 0.0f;
    for (int c = threadIdx.x; c < ATT; c += 256)
        acc += bh[c] * wt[c];
    s[threadIdx.x] = acc;
    __syncthreads();
    for (int st = 128; st > 0; st >>= 1) {
        if (threadIdx.x < st) s[threadIdx.x] += s[threadIdx.x + st];
        __syncthreads();
    }
    if (threadIdx.x == 0) c0out[0] = s[0] + bt[0];
}

// ---------------------------------------------------------------------------
// C[M,N] = A[M,K] * Bm[N,K]^T   (NT GEMM, full f32 via V_WMMA_F32_16X16X4_F32)
//
// Each wave owns a 16x64 output strip: 4 v8f accumulators, A fragment reused
// across the 4 WMMAs of every k-step (320 B of fragment traffic per WMMA vs
// 512 B for a lone 16x16 tile), and 4 independent WMMAs per step to hide
// matrix-pipe latency.
//
// Fragment layouts (ISA 7.12.2, 32-bit, wave32):
//   A 16x4 : lane -> (M = lane&15, K = k0 + 2*(lane>>4) + vgpr)
//   B 4x16 : lane -> (N = lane&15, K = k0 + 2*(lane>>4) + vgpr)
//   C 16x16: vgpr r -> (M = r + 8*(lane>>4), N = lane&15)
// M % 16 == 0, N % 64 == 0, K % 4 == 0 for both GEMMs -> EXEC all-ones
// around every WMMA (ISA requirement).
// ---------------------------------------------------------------------------
__global__ void gemm_nt_wmma_f32_kernel(const float* __restrict__ A,
                                        const float* __restrict__ Bm,
                                        float* __restrict__ C,
                                        int M, int N, int K) {
    const int lane = threadIdx.x & 31;
    const int wave = threadIdx.x >> 5;
    const int tile = blockIdx.x * (blockDim.x >> 5) + wave;
    const int tilesN = N >> 6;                   // 64-wide strips
    const int tm = tile / tilesN;
    const int tn = tile - tm * tilesN;
    if (tm * 16 >= M) return;                    // wave-uniform (never taken)

    const int l15  = lane & 15;
    const int half = lane >> 4;                  // 0: lanes 0-15, 1: 16-31
    const int m0 = tm << 4;
    const int n0 = tn << 6;

    const float* __restrict__ Arow = A  + (size_t)(m0 + l15) * K;
    const float* __restrict__ Brow = Bm + (size_t)(n0 + l15) * K;
    const size_t strideB = (size_t)16 * K;       // next 16-column strip

    v8f acc0 = {}, acc1 = {}, acc2 = {}, acc3 = {};
    #pragma unroll 2
    for (int k0 = 0; k0 < K; k0 += 4) {
        const int ka = k0 + half * 2;
        v2f a, b0, b1, b2, b3;
        a.x  = Arow[ka];                a.y  = Arow[ka + 1];
        b0.x = Brow[ka];                b0.y = Brow[ka + 1];
        b1.x = Brow[strideB + ka];      b1.y = Brow[strideB + ka + 1];
        b2.x = Brow[2 * strideB + ka];  b2.y = Brow[2 * strideB + ka + 1];
        b3.x = Brow[3 * strideB + ka];  b3.y = Brow[3 * strideB + ka + 1];
        acc0 = __builtin_amdgcn_wmma_f32_16x16x4_f32(false, a, false, b0,
                                                     (short)0, acc0, false, false);
        acc1 = __builtin_amdgcn_wmma_f32_16x16x4_f32(false, a, false, b1,
                                                     (short)0, acc1, false, false);
        acc2 = __builtin_amdgcn_wmma_f32_16x16x4_f32(false, a, false, b2,
                                                     (short)0, acc2, false, false);
        acc3 = __builtin_amdgcn_wmma_f32_16x16x4_f32(false, a, false, b3,
                                                     (short)0, acc3, false, false);
    }

    float* __restrict__ Crow = C + (size_t)(m0 + half * 8) * N + n0 + l15;
    #pragma unroll
    for (int r = 0; r < 8; ++r) {
        Crow[(size_t)r * N + 0]  = acc0[r];
        Crow[(size_t)r * N + 16] = acc1[r];
        Crow[(size_t)r * N + 32] = acc2[r];
        Crow[(size_t)r * N + 48] = acc3[r];
    }
}

// ---------------------------------------------------------------------------
// alpha[b,l,p] = sum_a tanh(p1[b,p,a] * p2[b,l,a]) * v[a] + c0
//
// One block per (b,l). Each lane owns the contiguous slice a in
// [lane*32, lane*32+32) and caches p2[b,l,slice] and v[slice] in REGISTERS
// (64 VGPRs) -- the inner loop touches no LDS at all. Per 4 elements:
// one global_load_b128 (L2-resident p1), 4x v_mul, 4x v_tanh_f32 (TRANS,
// co-executes), 4x v_fmac. Wave32 shuffle reduction finishes each p.
// ---------------------------------------------------------------------------
__global__ void attn_fused_kernel(const float* __restrict__ p1,
                                  const float* __restrict__ p2,
                                  const float* __restrict__ v,
                                  const float* __restrict__ c0p,
                                  float* __restrict__ out) {
    const int bl = blockIdx.x;                   // 0 .. B*L-1
    const int b  = bl / LL;
    const int lane  = threadIdx.x & 31;
    const int wave  = threadIdx.x >> 5;
    const int nwave = blockDim.x >> 5;

    // Cache p2 row slice and v slice in registers (fully unrolled indexing).
    float rp2[32], rv[32];
    {
        const float4* __restrict__ p2s = (const float4*)(p2 + (size_t)bl * ATT + lane * 32);
        const float4* __restrict__ vs  = (const float4*)(v + lane * 32);
        #pragma unroll
        for (int i = 0; i < 8; ++i) {
            float4 t2 = p2s[i];
            float4 tv = vs[i];
            rp2[4*i+0] = t2.x; rp2[4*i+1] = t2.y; rp2[4*i+2] = t2.z; rp2[4*i+3] = t2.w;
            rv [4*i+0] = tv.x; rv [4*i+1] = tv.y; rv [4*i+2] = tv.z; rv [4*i+3] = tv.w;
        }
    }
    const float c0 = c0p[0];
    const float* __restrict__ p1base = p1 + (size_t)b * PP * ATT + lane * 32;

    for (int p = wave; p < PP; p += nwave) {
        const float4* __restrict__ p1s = (const float4*)(p1base + (size_t)p * ATT);
        if (p + nwave < PP)                       // prefetch next row for this wave
            __builtin_prefetch(p1base + (size_t)(p + nwave) * ATT, 0, 0);
        float acc = 0.0f;
        #pragma unroll
        for (int i = 0; i < 8; ++i) {
            float4 x = p1s[i];
            acc += fast_tanh(x.x * rp2[4*i+0]) * rv[4*i+0];
            acc += fast_tanh(x.y * rp2[4*i+1]) * rv[4*i+1];
            acc += fast_tanh(x.z * rp2[4*i+2]) * rv[4*i+2];
            acc += fast_tanh(x.w * rp2[4*i+3]) * rv[4*i+3];
        }
        #pragma unroll
        for (int off = 16; off > 0; off >>= 1)
            acc += __shfl_down(acc, off, 32);
        if (lane == 0)
            out[(size_t)bl * PP + p] = acc + c0;
    }
}

// ---------------------------------------------------------------------------
// Launcher
// ---------------------------------------------------------------------------
extern "C" void kernel_launch(void* const* d_in, const int* in_sizes, int n_in,
                              void* d_out, int out_size, void* d_ws, size_t ws_size,
                              hipStream_t stream) {
    const float* x1 = (const float*)d_in[0];   // (B,P,DIM1)
    const float* x2 = (const float*)d_in[1];   // (B,L,DIM2)
    const float* W1 = (const float*)d_in[2];   // (ATT,DIM1)
    const float* W2 = (const float*)d_in[3];   // (ATT,DIM2)
    const float* Wh = (const float*)d_in[4];   // (ATT,ATT)
    const float* bh = (const float*)d_in[5];   // (ATT,)
    const float* wt = (const float*)d_in[6];   // (ATT,)
    const float* bt = (const float*)d_in[7];   // scalar
    float* out = (float*)d_out;                // (B,L,P)

    // Workspace layout (all f32): p1 | p2 | v | c0   (~9.05 MB, L2-resident)
    float* p1 = (float*)d_ws;                      // B*P x ATT
    float* p2 = p1 + (size_t)BB * PP * ATT;        // B*L x ATT
    float* vv = p2 + (size_t)BB * LL * ATT;        // ATT
    float* c0 = vv + ATT;                          // 1

    // v = Wh^T * wt ; c0 = bh.wt + bt   (collapses the 263-GFLOP blpc GEMM)
    precompute_v_kernel<<<ATT / 256, 256, 0, stream>>>(Wh, wt, vv);
    precompute_c0_kernel<<<1, 256, 0, stream>>>(bh, wt, bt, c0);

    // p1 = x1 * W1^T : M=1568, N=1024, K=2048  -> 98*16 = 1568 wave-strips
    {
        const int M = BB * PP, N = ATT, K = DIM1;
        const int strips = (M / 16) * (N / 64);
        gemm_nt_wmma_f32_kernel<<<strips / 8, 256, 0, stream>>>(x1, W1, p1, M, N, K);
    }
    // p2 = x2 * W2^T : M=640, N=1024, K=300   -> 40*16 = 640 wave-strips
    {
        const int M = BB * LL, N = ATT, K = DIM2;
        const int strips = (M / 16) * (N / 64);
        gemm_nt_wmma_f32_kernel<<<strips / 8, 256, 0, stream>>>(x2, W2, p2, M, N, K);
    }

    // Fused tanh-bilinear contraction -> alpha (B,L,P)
    attn_fused_kernel<<<BB * LL, 256, 0, stream>>>(p1, p2, vv, c0, out);
}